// StaticRecurrentEntNet_75350906241661
// MI455X (gfx1250) — compile-verified
//
#include <hip/hip_runtime.h>
#include <hip/hip_bf16.h>

// Problem constants (from reference)
#define BATCH 256
#define SMAX  30
#define LLEN  32
#define DDIM  128
#define ENUM  64
#define EPS_L2 1e-12f

typedef __attribute__((ext_vector_type(2))) float v2f;
typedef __attribute__((ext_vector_type(8))) float v8f;

// D = A(16x4 f32) * B(4x16 f32) + C(16x16 f32), wave32, CDNA5
__device__ __forceinline__ v8f wmma_f32_16x16x4(v2f a, v2f b, v8f c) {
    return __builtin_amdgcn_wmma_f32_16x16x4_f32(
        /*neg_a=*/false, a, /*neg_b=*/false, b,
        /*c_mod=*/(short)0, c, /*reuse_a=*/false, /*reuse_b=*/false);
}

// ---------------------------------------------------------------------------
// Kernel 1: enc[b,s,d] = sum_l prgrph[b,s,l,d]
// grid = B*S blocks of 128 threads (thread = d), coalesced over d.
// ---------------------------------------------------------------------------
__global__ void k_sent_encode(const float* __restrict__ x, float* __restrict__ enc) {
    const int bs = blockIdx.x;          // b*S + s
    const int d  = threadIdx.x;         // 0..127
    const float* p = x + (size_t)bs * LLEN * DDIM + d;
    float acc = 0.f;
#pragma unroll
    for (int l = 0; l < LLEN; ++l) acc += p[l * DDIM];
    enc[(size_t)bs * DDIM + d] = acc;
}

// ---------------------------------------------------------------------------
// Kernel 2: generic  out[r,:] = A[r,:] @ M   (A: R x 128, M: 128 x 128)
// block = 256 threads (8 waves), each block = one 16-row M-tile, wave w = N-tile w.
// Both operands staged in LDS; K-loop of 32 x V_WMMA_F32_16X16X4_F32 per tile.
// ---------------------------------------------------------------------------
#define AP 132  // padded LDS pitch (132 % 64 = 4): conflict-free lane-strided reads
__global__ void k_gemm128(const float* __restrict__ A, const float* __restrict__ M,
                          float* __restrict__ out) {
    extern __shared__ float smem[];
    float* aT = smem;              // 16 * AP
    float* mB = aT + 16 * AP;      // 128 * 128
    const int tid = threadIdx.x;
    const int m0  = blockIdx.x * 16;

    // stage A tile (rows padded to AP) and the whole weight matrix, float4 traffic
    for (int idx = tid; idx < 16 * (DDIM / 4); idx += 256) {
        int r = idx >> 5, c4 = idx & 31;
        ((float4*)(aT + r * AP))[c4] =
            ((const float4*)(A + (size_t)(m0 + r) * DDIM))[c4];
    }
    {
        const float4* M4 = (const float4*)M;
        float4* mB4 = (float4*)mB;
        for (int idx = tid; idx < (DDIM * DDIM) / 4; idx += 256) mB4[idx] = M4[idx];
    }
    __syncthreads();

    const int wave = tid >> 5, lane = tid & 31;
    const int n0   = wave * 16;
    const int laneN = lane & 15;
    const int hi    = lane >> 4;     // 0/1
    const int koff  = hi * 2;        // A/B vgpr0 holds K=0 (lo lanes) / K=2 (hi lanes)

    v8f acc = {};
#pragma unroll
    for (int kk = 0; kk < 32; ++kk) {
        const int k = 4 * kk + koff;
        v2f a, b;
        a.x = aT[laneN * AP + k];
        a.y = aT[laneN * AP + k + 1];
        b.x = mB[k * DDIM + n0 + laneN];
        b.y = mB[(k + 1) * DDIM + n0 + laneN];
        acc = wmma_f32_16x16x4(a, b, acc);
    }
#pragma unroll
    for (int r = 0; r < 8; ++r) {
        out[(size_t)(m0 + r + hi * 8) * DDIM + n0 + laneN] = acc[r];
    }
}

// ---------------------------------------------------------------------------
// Kernel 3: ek[b,s,e] = dot(enc[b,s,:], keys[b,e,:])   (gate bias term)
// grid = B, block = 256 (8 waves). Tiles: S(pad 32)/16 x E/16 = 2 x 4 = 8 tiles.
// ---------------------------------------------------------------------------
__global__ void k_enc_dot_keys(const float* __restrict__ enc, const float* __restrict__ keys,
                               float* __restrict__ ek) {
    __shared__ float encT[32 * AP];    // padded sentence rows (30 valid + 2 zero)
    __shared__ float keyT[64 * AP];    // keys[b], padded (B-operand reads stride AP)
    const int tid = threadIdx.x;
    const int b   = blockIdx.x;

    for (int idx = tid; idx < 32 * (DDIM / 4); idx += 256) {
        int r = idx >> 5, c4 = idx & 31;
        float4 v = make_float4(0.f, 0.f, 0.f, 0.f);
        if (r < SMAX) v = ((const float4*)(enc + ((size_t)b * SMAX + r) * DDIM))[c4];
        ((float4*)(encT + r * AP))[c4] = v;
    }
    for (int idx = tid; idx < 64 * (DDIM / 4); idx += 256) {
        int r = idx >> 5, c4 = idx & 31;
        ((float4*)(keyT + r * AP))[c4] =
            ((const float4*)(keys + ((size_t)b * ENUM + r) * DDIM))[c4];
    }
    __syncthreads();

    const int wave = tid >> 5, lane = tid & 31;
    const int m0 = (wave >> 2) * 16;   // sentence tile
    const int n0 = (wave & 3) * 16;    // entity tile
    const int laneN = lane & 15;
    const int hi = lane >> 4, koff = hi * 2;

    v8f acc = {};
#pragma unroll
    for (int kk = 0; kk < 32; ++kk) {
        const int k = 4 * kk + koff;
        v2f a, bb;
        a.x  = encT[(m0 + laneN) * AP + k];
        a.y  = encT[(m0 + laneN) * AP + k + 1];
        bb.x = keyT[(n0 + laneN) * AP + k];      // B[k][n] = keys[n][k]
        bb.y = keyT[(n0 + laneN) * AP + k + 1];
        acc = wmma_f32_16x16x4(a, bb, acc);
    }
#pragma unroll
    for (int r = 0; r < 8; ++r) {
        const int s = m0 + r + hi * 8;
        if (s < SMAX) ek[((size_t)b * SMAX + s) * ENUM + n0 + laneN] = acc[r];
    }
}

// ---------------------------------------------------------------------------
// Kernel 4: the recurrence. One workgroup per batch element, 30 sequential
// steps, all hot state (h, kV, U) resident in LDS. Per step: VALU gate dots,
// WMMA h@U (8 waves x 4 N-tiles x 32 K-steps, A fragment reused across the
// 4 tiles), masked gated update, l2 norm. Next step's rows are prefetched
// (global_prefetch_b8) while the WMMA chain runs.
// ---------------------------------------------------------------------------
#define HP 132  // padded pitch for h in LDS (A-operand reads stride HP)
__global__ void k_entnet_scan(const float* __restrict__ U,
                              const float* __restrict__ kV,
                              const float* __restrict__ enc,
                              const float* __restrict__ encW,
                              const float* __restrict__ ek,
                              const unsigned char* __restrict__ mask,
                              float* __restrict__ out) {
    extern __shared__ float smem[];
    float* hBuf    = smem;                 // ENUM * HP
    float* kvB     = hBuf + ENUM * HP;     // ENUM * DDIM
    float* uB      = kvB + ENUM * DDIM;    // DDIM * DDIM
    float* encRow  = uB + DDIM * DDIM;     // DDIM
    float* encWRow = encRow + DDIM;        // DDIM
    float* ekRow   = encWRow + DDIM;       // ENUM
    float* gatesB  = ekRow + ENUM;         // ENUM
    float* scaleB  = gatesB + ENUM;        // ENUM
    float* partB   = scaleB + ENUM;        // 256

    const int tid = threadIdx.x;
    const int b   = blockIdx.x;

    // ---- resident state init (vectorized) ----
    for (int idx = tid; idx < ENUM * HP; idx += 256) hBuf[idx] = 0.f;
    {
        const float4* kv4 = (const float4*)(kV + (size_t)b * ENUM * DDIM);
        const float4* U4  = (const float4*)U;
        float4* kvB4 = (float4*)kvB;
        float4* uB4  = (float4*)uB;
        for (int idx = tid; idx < (ENUM * DDIM) / 4; idx += 256) kvB4[idx] = kv4[idx];
        for (int idx = tid; idx < (DDIM * DDIM) / 4; idx += 256) uB4[idx] = U4[idx];
    }
    __syncthreads();

    const int wave = tid >> 5, lane = tid & 31;
    const int mt = wave >> 1;              // 0..3
    const int m0 = mt * 16;
    const int ntBase = (wave & 1) * 4;     // 0 or 4
    const int laneN = lane & 15;
    const int hi = lane >> 4, koff = hi * 2;
    const int eQ = tid >> 2, q32 = (tid & 3) * 32;   // for 4-way row reductions

#pragma unroll 1
    for (int s = 0; s < SMAX; ++s) {
        // per-step rows
        if (tid < DDIM) {
            encRow[tid]  = enc [((size_t)b * SMAX + s) * DDIM + tid];
            encWRow[tid] = encW[((size_t)b * SMAX + s) * DDIM + tid];
        }
        if (tid < ENUM) ekRow[tid] = ek[((size_t)b * SMAX + s) * ENUM + tid];
        const float actf = mask[b * SMAX + s] ? 1.f : 0.f;
        // prefetch next step's rows into cache while this step computes
        if (s + 1 < SMAX) {
            const size_t nb = ((size_t)b * SMAX + s + 1);
            if (tid < 4) {
                __builtin_prefetch(enc  + nb * DDIM + tid * 32, 0, 1);
                __builtin_prefetch(encW + nb * DDIM + tid * 32, 0, 1);
            }
            if (tid == 4) __builtin_prefetch(ek + nb * ENUM, 0, 1);
        }
        __syncthreads();

        // gates[e] = sigmoid(dot(enc, h[e]) + dot(enc, k[e]))
        {
            float p = 0.f;
#pragma unroll 8
            for (int j = 0; j < 32; ++j) p += encRow[q32 + j] * hBuf[eQ * HP + q32 + j];
            partB[tid] = p;
        }
        __syncthreads();
        if (tid < ENUM) {
            float g = partB[4 * tid] + partB[4 * tid + 1] + partB[4 * tid + 2]
                    + partB[4 * tid + 3] + ekRow[tid];
            gatesB[tid] = 1.f / (1.f + __expf(-g));
        }
        __syncthreads();

        // h_tilda = relu(h @ U + kV + encW); 4 N-tiles per wave share the A fragment
        v8f acc[4];
#pragma unroll
        for (int t4 = 0; t4 < 4; ++t4) {
            const int n0 = (ntBase + t4) * 16;
#pragma unroll
            for (int r = 0; r < 8; ++r)
                acc[t4][r] = kvB[(m0 + r + hi * 8) * DDIM + n0 + laneN]
                           + encWRow[n0 + laneN];
        }
#pragma unroll
        for (int kk = 0; kk < 32; ++kk) {
            const int k = 4 * kk + koff;
            v2f a;
            a.x = hBuf[(m0 + laneN) * HP + k];
            a.y = hBuf[(m0 + laneN) * HP + k + 1];
#pragma unroll
            for (int t4 = 0; t4 < 4; ++t4) {
                const int n0 = (ntBase + t4) * 16;
                v2f bb;
                bb.x = uB[k * DDIM + n0 + laneN];
                bb.y = uB[(k + 1) * DDIM + n0 + laneN];
                acc[t4] = wmma_f32_16x16x4(a, bb, acc[t4]);
            }
        }
        // masked gated update (reads old h + gates, so before the write barrier)
#pragma unroll
        for (int t4 = 0; t4 < 4; ++t4) {
            const int n0 = (ntBase + t4) * 16;
#pragma unroll
            for (int r = 0; r < 8; ++r) {
                const int e = m0 + r + hi * 8;
                const float ht = fmaxf(acc[t4][r], 0.f);
                acc[t4][r] = hBuf[e * HP + n0 + laneN] + actf * gatesB[e] * ht;
            }
        }
        __syncthreads();   // all waves done reading old h
#pragma unroll
        for (int t4 = 0; t4 < 4; ++t4)
#pragma unroll
            for (int r = 0; r < 8; ++r)
                hBuf[(m0 + r + hi * 8) * HP + (ntBase + t4) * 16 + laneN] = acc[t4][r];
        __syncthreads();

        // l2 normalize rows of h
        {
            float p = 0.f;
#pragma unroll 8
            for (int j = 0; j < 32; ++j) {
                float v = hBuf[eQ * HP + q32 + j];
                p += v * v;
            }
            partB[tid] = p;
        }
        __syncthreads();
        if (tid < ENUM) {
            float ss = partB[4 * tid] + partB[4 * tid + 1] + partB[4 * tid + 2] + partB[4 * tid + 3];
            scaleB[tid] = rsqrtf(fmaxf(ss, EPS_L2));
        }
        __syncthreads();
#pragma unroll 8
        for (int j = 0; j < 32; ++j) hBuf[eQ * HP + q32 + j] *= scaleB[eQ];
        __syncthreads();
    }

    // write final hiddens (float4, padded-pitch safe: 132*4B is 16B aligned)
    {
        float4* out4 = (float4*)(out + (size_t)b * ENUM * DDIM);
        for (int idx = tid; idx < ENUM * (DDIM / 4); idx += 256) {
            int e = idx >> 5, c4 = idx & 31;
            out4[idx] = ((const float4*)(hBuf + e * HP))[c4];
        }
    }
}

// ---------------------------------------------------------------------------
extern "C" void kernel_launch(void* const* d_in, const int* in_sizes, int n_in,
                              void* d_out, int out_size, void* d_ws, size_t ws_size,
                              hipStream_t stream) {
    (void)in_sizes; (void)n_in; (void)out_size; (void)ws_size;
    const float* entity_keys = (const float*)d_in[0];          // [B,E,D]
    const float* prgrph      = (const float*)d_in[1];          // [B,S,L,D]
    const unsigned char* msk = (const unsigned char*)d_in[2];  // [B,S] bool
    const float* U = (const float*)d_in[3];                    // [D,D]
    const float* V = (const float*)d_in[4];
    const float* W = (const float*)d_in[5];
    float* out = (float*)d_out;

    // workspace layout (floats)
    float* ws   = (float*)d_ws;
    float* enc  = ws;                                  // B*S*D
    float* encW = enc  + (size_t)BATCH * SMAX * DDIM;  // B*S*D
    float* kV   = encW + (size_t)BATCH * SMAX * DDIM;  // B*E*D
    float* ek   = kV   + (size_t)BATCH * ENUM * DDIM;  // B*S*E

    // 1. bag-of-words sentence encodings
    k_sent_encode<<<BATCH * SMAX, DDIM, 0, stream>>>(prgrph, enc);

    // 2. hoisted GEMMs: encW = enc @ W ; kV = keys @ V (weights staged in LDS)
    const size_t gemmSmem = (16 * AP + DDIM * DDIM) * sizeof(float);
    k_gemm128<<<(BATCH * SMAX) / 16, 256, gemmSmem, stream>>>(enc, W, encW);
    k_gemm128<<<(BATCH * ENUM) / 16, 256, gemmSmem, stream>>>(entity_keys, V, kV);

    // 3. gate bias: ek[b,s,e] = enc[b,s] . keys[b,e]
    k_enc_dot_keys<<<BATCH, 256, 0, stream>>>(enc, entity_keys, ek);

    // 4. sequential scan, one workgroup per batch element, state in LDS
    const size_t scanSmem =
        (ENUM * HP + ENUM * DDIM + DDIM * DDIM + 2 * DDIM + 3 * ENUM + 256) * sizeof(float);
    k_entnet_scan<<<BATCH, 256, scanSmem, stream>>>(U, kV, enc, encW, ek, msk, out);
}